// AttnGCN_81767587381711
// MI455X (gfx1250) — compile-verified
//
#include <hip/hip_runtime.h>
#include <hip/hip_bf16.h>

// ---------------- problem constants ----------------
#define NN      200000
#define NE      6400000
#define F_INN   20
#define F_OUTN  10
#define NEG_SLOPE 0.2f
#define BN_EPS  1e-5f

#define NB_STAT 512
#define NB_EW   512

// ---------------- workspace layout (float offsets) ----------------
#define WS_XP   0                         // N*10
#define WS_AS   2000000                   // N
#define WS_AD   2200000                   // N
#define WS_M    2400000                   // N  (segment max)
#define WS_DEN  2600000                   // N  (segment sum)
#define WS_OUT  2800000                   // N*10 (aggregated messages)
#define WS_PS   4800000                   // NB_STAT*40 bn partials
#define WS_PE   (WS_PS + NB_STAT * 40)    // NB_EW edge-weight partials
#define WS_SCAL (WS_PE + NB_EW)           // 128 scalars:
// scal[0..19]=mu  [20..39]=scale=rsqrt(var+eps)*bn_w  [40..59]=bn_b
// scal[60]=mean_ew  scal[61]=we_dot
#define WS_BTP  (WS_SCAL + 128)           // 20x16 padded B for projection
#define WS_BTM  (WS_BTP + 320)            // 3 x 12x16 padded B for MLP layers
#define WS_BTB  (WS_BTM + 576)            // 3 x 16 padded biases for MLP

typedef float v2f __attribute__((ext_vector_type(2)));
typedef float v8f __attribute__((ext_vector_type(8)));

__device__ __forceinline__ v8f wmma_f32_k4(v2f a, v2f b, v8f c) {
  // D = A(16x4) * B(4x16) + C(16x16), all f32, wave32
  return __builtin_amdgcn_wmma_f32_16x16x4_f32(false, a, false, b, (short)0, c,
                                               false, false);
}

// float atomic max via signed/unsigned integer atomics (IEEE order trick)
__device__ __forceinline__ void atomicMaxF(float* addr, float val) {
  int iv = __float_as_int(val);
  if (iv >= 0) atomicMax((int*)addr, iv);
  else         atomicMin((unsigned int*)addr, (unsigned int)iv);
}

// ---------------- init: zero accumulators, m = -inf ----------------
__global__ void k_init(float* __restrict__ out_acc, float* __restrict__ m,
                       float* __restrict__ den) {
  int i = blockIdx.x * 256 + threadIdx.x;
  if (i < NN * F_OUTN) out_acc[i] = 0.0f;
  if (i < NN) { m[i] = -INFINITY; den[i] = 0.0f; }
}

// ---------------- BN stats: per-block partial sums (deterministic) ----------
__global__ void k_bn_partial(const float* __restrict__ h, float* __restrict__ ps) {
  __shared__ float red[256][41];
  float ls[F_INN], lq[F_INN];
#pragma unroll
  for (int f = 0; f < F_INN; ++f) { ls[f] = 0.0f; lq[f] = 0.0f; }
  for (int n = blockIdx.x * 256 + threadIdx.x; n < NN; n += NB_STAT * 256) {
#pragma unroll
    for (int f = 0; f < F_INN; ++f) {
      float v = h[n * F_INN + f];
      ls[f] += v; lq[f] += v * v;
    }
  }
#pragma unroll
  for (int f = 0; f < F_INN; ++f) {
    red[threadIdx.x][f]         = ls[f];
    red[threadIdx.x][F_INN + f] = lq[f];
  }
  __syncthreads();
  if (threadIdx.x < 40) {
    float s = 0.0f;
    for (int t = 0; t < 256; ++t) s += red[t][threadIdx.x];
    ps[blockIdx.x * 40 + threadIdx.x] = s;
  }
}

// ---------------- edge-weight mean partials ----------------
__global__ void k_ew_partial(const float* __restrict__ ew, float* __restrict__ pe) {
  __shared__ float red[256];
  float s = 0.0f;
  for (int e = blockIdx.x * 256 + threadIdx.x; e < NE; e += NB_EW * 256)
    s += ew[e];
  red[threadIdx.x] = s;
  __syncthreads();
  for (int off = 128; off > 0; off >>= 1) {
    if (threadIdx.x < off) red[threadIdx.x] += red[threadIdx.x + off];
    __syncthreads();
  }
  if (threadIdx.x == 0) pe[blockIdx.x] = red[0];
}

// ------- finalize scalars + build padded branchless WMMA B tables ----------
__global__ void k_finalize(const float* __restrict__ ps, const float* __restrict__ pe,
                           const float* __restrict__ bn_w, const float* __restrict__ bn_b,
                           const float* __restrict__ W,
                           const float* __restrict__ att_src,
                           const float* __restrict__ att_dst,
                           const float* __restrict__ W_edge,
                           const float* __restrict__ att_edge,
                           const float* __restrict__ w1, const float* __restrict__ b1,
                           const float* __restrict__ w2, const float* __restrict__ b2,
                           const float* __restrict__ w3, const float* __restrict__ b3,
                           float* __restrict__ scal, float* __restrict__ btp,
                           float* __restrict__ btm, float* __restrict__ btb) {
  const int t = threadIdx.x;
  if (t < F_INN) {
    // BN batch stats (biased variance), fused with affine params
    float S = 0.0f, Q = 0.0f;
    for (int b = 0; b < NB_STAT; ++b) {
      S += ps[b * 40 + t];
      Q += ps[b * 40 + F_INN + t];
    }
    float mu  = S / (float)NN;
    float var = Q / (float)NN - mu * mu;
    scal[t]      = mu;
    scal[20 + t] = rsqrtf(var + BN_EPS) * bn_w[t];
    scal[40 + t] = bn_b[t];
    // Projection B row t (K index): cols 0..9 = W^T, 10 = W^T att_src, 11 = W^T att_dst
    float sd = 0.0f, dd = 0.0f;
#pragma unroll
    for (int n = 0; n < F_OUTN; ++n) {
      float w = W[n * F_INN + t];
      sd += w * att_src[n];
      dd += w * att_dst[n];
    }
#pragma unroll
    for (int c = 0; c < 16; ++c) {
      float v = 0.0f;
      if (c < F_OUTN)      v = W[c * F_INN + t];
      else if (c == 10)    v = sd;
      else if (c == 11)    v = dd;
      btp[t * 16 + c] = v;
    }
  } else if (t == F_INN) {
    float S = 0.0f;
    for (int b = 0; b < NB_EW; ++b) S += pe[b];
    scal[60] = S / (float)NE;          // mean edge weight (self-loop attr)
    float wd = 0.0f;
#pragma unroll
    for (int n = 0; n < F_OUTN; ++n) wd += W_edge[n] * att_edge[n];
    scal[61] = wd;                      // att_edge . W_edge
  }
  // MLP padded B tables: 3 layers x 12(K) x 16(N)
  if (t >= 64 && t < 64 + 36) {
    const int L = (t - 64) / 12, k = (t - 64) % 12;
    const float* w = (L == 0) ? w1 : (L == 1) ? w2 : w3;
#pragma unroll
    for (int c = 0; c < 16; ++c)
      btm[(L * 12 + k) * 16 + c] = (k < F_OUTN && c < F_OUTN) ? w[c * F_OUTN + k] : 0.0f;
  }
  // MLP padded bias rows: 3 x 16
  if (t >= 100 && t < 103) {
    const int L = t - 100;
    const float* b = (L == 0) ? b1 : (L == 1) ? b2 : b3;
#pragma unroll
    for (int c = 0; c < 16; ++c)
      btb[L * 16 + c] = (c < F_OUTN) ? b[c] : 0.0f;
  }
}

// ---------------- fused BN + projection via WMMA f32 16x16x4 ----------------
// One wave = one 16-node tile. B fragments come from the padded table -> no
// divergent branches between consecutive v_wmma ops.
__global__ void k_project(const float* __restrict__ h, const float* __restrict__ scal,
                          const float* __restrict__ btp,
                          float* __restrict__ xp, float* __restrict__ a_s,
                          float* __restrict__ a_d) {
  const int wave = (blockIdx.x * blockDim.x + threadIdx.x) >> 5;
  if (wave * 16 >= NN) return;                 // wave-uniform: EXEC stays all-1s
  const int lane = threadIdx.x & 31;
  const int hh   = lane >> 4;                  // lane half selects K sub-pair
  const int mn   = lane & 15;                  // A row / B,C col
  const int node = wave * 16 + mn;

  const float* mu = scal;
  const float* sc = scal + 20;
  const float* sh = scal + 40;

  v8f acc = {};
#pragma unroll
  for (int kk = 0; kk < F_INN; kk += 4) {
    const int k0 = kk + hh * 2;
    v2f a, b;
    a.x = (h[node * F_INN + k0]     - mu[k0])     * sc[k0]     + sh[k0];
    a.y = (h[node * F_INN + k0 + 1] - mu[k0 + 1]) * sc[k0 + 1] + sh[k0 + 1];
    b.x = btp[k0 * 16 + mn];                   // branchless B fragment
    b.y = btp[(k0 + 1) * 16 + mn];
    acc = wmma_f32_k4(a, b, acc);
  }
  // C layout: VGPR i -> row (i + hh*8), col mn
#pragma unroll
  for (int i = 0; i < 8; ++i) {
    const int on = wave * 16 + i + hh * 8;
    const float v = acc[i];
    if (mn < F_OUTN)      xp[on * F_OUTN + mn] = v;
    else if (mn == 10)    a_s[on] = v;
    else if (mn == 11)    a_d[on] = v;
  }
}

// ---------------- edge logit helper ----------------
__device__ __forceinline__ float edge_logit(int idx, const int* __restrict__ si,
                                            const int* __restrict__ di,
                                            const float* __restrict__ ew,
                                            const float* __restrict__ a_s,
                                            const float* __restrict__ a_d,
                                            const float* __restrict__ scal,
                                            int& s, int& d) {
  float ea;
  if (idx < NE) { s = si[idx]; d = di[idx]; ea = ew[idx]; }
  else          { s = d = idx - NE;         ea = scal[60]; }   // self loop, mean attr
  float x = a_s[s] + a_d[d] + scal[61] * ea;
  return x > 0.0f ? x : NEG_SLOPE * x;                          // leaky relu
}

// ---------------- pass A: segment max ----------------
__global__ void k_edge_max(const int* __restrict__ si, const int* __restrict__ di,
                           const float* __restrict__ ew, const float* __restrict__ a_s,
                           const float* __restrict__ a_d, const float* __restrict__ scal,
                           float* __restrict__ m) {
  int idx = blockIdx.x * 256 + threadIdx.x;
  if (idx >= NE + NN) return;
  int s, d;
  float x = edge_logit(idx, si, di, ew, a_s, a_d, scal, s, d);
  atomicMaxF(&m[d], x);
}

// ---------------- pass B: exp & segment sum ----------------
__global__ void k_edge_expsum(const int* __restrict__ si, const int* __restrict__ di,
                              const float* __restrict__ ew, const float* __restrict__ a_s,
                              const float* __restrict__ a_d, const float* __restrict__ scal,
                              const float* __restrict__ m, float* __restrict__ den) {
  int idx = blockIdx.x * 256 + threadIdx.x;
  if (idx >= NE + NN) return;
  int s, d;
  float x = edge_logit(idx, si, di, ew, a_s, a_d, scal, s, d);
  atomicAdd(&den[d], expf(x - m[d]));
}

// ---------------- pass C: weighted scatter of messages ----------------
__global__ void k_edge_scatter(const int* __restrict__ si, const int* __restrict__ di,
                               const float* __restrict__ ew, const float* __restrict__ a_s,
                               const float* __restrict__ a_d, const float* __restrict__ scal,
                               const float* __restrict__ m, const float* __restrict__ den,
                               const float* __restrict__ xp, float* __restrict__ out_acc) {
  int idx = blockIdx.x * 256 + threadIdx.x;
  if (idx >= NE + NN) return;
  if (idx + 4096 < NE) {                       // stream-ahead hint -> global_prefetch_b8
    __builtin_prefetch(si + idx + 4096, 0, 1);
    __builtin_prefetch(di + idx + 4096, 0, 1);
  }
  int s, d;
  float x = edge_logit(idx, si, di, ew, a_s, a_d, scal, s, d);
  float alpha = expf(x - m[d]) / den[d];
#pragma unroll
  for (int f = 0; f < F_OUTN; ++f)
    atomicAdd(&out_acc[d * F_OUTN + f], xp[s * F_OUTN + f] * alpha);
}

// ---------------- embeddings = relu(out + bias) ----------------
__global__ void k_embed(const float* __restrict__ out_acc, const float* __restrict__ bias,
                        float* __restrict__ emb) {
  int i = blockIdx.x * 256 + threadIdx.x;
  if (i >= NN * F_OUTN) return;
  emb[i] = fmaxf(out_acc[i] + bias[i % F_OUTN], 0.0f);
}

// ------- MLP: 3 chained 10x10 layers via WMMA, LDS transpose, table B -------
__global__ void k_mlp(const float* __restrict__ out_acc, const float* __restrict__ bias,
                      const float* __restrict__ btm, const float* __restrict__ btb,
                      float* __restrict__ y) {
  __shared__ float smem[8][16][18];            // per-wave 16x16 tile, padded
  const int wv   = threadIdx.x >> 5;
  const int lane = threadIdx.x & 31;
  const int tile = blockIdx.x * 8 + wv;
  const bool active = (tile * 16 < NN);        // wave-uniform
  const int t    = active ? tile : 0;
  const int hh   = lane >> 4;
  const int mn   = lane & 15;
  const int node = t * 16 + mn;
  const int k0 = hh * 2, k1 = 4 + hh * 2, k2 = 8 + hh * 2;

  // A = GAT output (raw, + bias), K padded 10 -> 12
  v2f a0, a1, a2;
  a0.x = out_acc[node * F_OUTN + k0]     + bias[k0];
  a0.y = out_acc[node * F_OUTN + k0 + 1] + bias[k0 + 1];
  a1.x = out_acc[node * F_OUTN + k1]     + bias[k1];
  a1.y = out_acc[node * F_OUTN + k1 + 1] + bias[k1 + 1];
  a2.x = (k2     < F_OUTN) ? out_acc[node * F_OUTN + k2]     + bias[k2]     : 0.0f;
  a2.y = (k2 + 1 < F_OUTN) ? out_acc[node * F_OUTN + k2 + 1] + bias[k2 + 1] : 0.0f;

#pragma unroll
  for (int L = 0; L < 3; ++L) {
    const float* bt = btm + L * 12 * 16;
    v2f b0, b1, b2;                            // branchless B fragments
    b0.x = bt[k0 * 16 + mn];        b0.y = bt[(k0 + 1) * 16 + mn];
    b1.x = bt[k1 * 16 + mn];        b1.y = bt[(k1 + 1) * 16 + mn];
    b2.x = bt[k2 * 16 + mn];        b2.y = bt[(k2 + 1) * 16 + mn];
    v8f acc = {};
    acc = wmma_f32_k4(a0, b0, acc);
    acc = wmma_f32_k4(a1, b1, acc);
    acc = wmma_f32_k4(a2, b2, acc);
    const float bb = btb[L * 16 + mn];         // padded bias (0 for cols 10..15)
    if (L < 2) {
      // relu, park in LDS row-major, re-read as A fragments (layout transpose)
#pragma unroll
      for (int i = 0; i < 8; ++i)
        smem[wv][i + hh * 8][mn] = fmaxf(acc[i] + bb, 0.0f);
      __syncthreads();
      a0.x = smem[wv][mn][k0];     a0.y = smem[wv][mn][k0 + 1];
      a1.x = smem[wv][mn][k1];     a1.y = smem[wv][mn][k1 + 1];
      a2.x = smem[wv][mn][k2];     a2.y = smem[wv][mn][k2 + 1];   // cols 10,11 are 0
      __syncthreads();
    } else if (active && mn < F_OUTN) {
#pragma unroll
      for (int i = 0; i < 8; ++i)
        y[(t * 16 + i + hh * 8) * F_OUTN + mn] = acc[i] + bb;
    }
  }
}

// ---------------- launch ----------------
extern "C" void kernel_launch(void* const* d_in, const int* in_sizes, int n_in,
                              void* d_out, int out_size, void* d_ws, size_t ws_size,
                              hipStream_t stream) {
  const float* h        = (const float*)d_in[0];
  const int*   ei       = (const int*)  d_in[1];   // [2,E] row-major
  const float* ew       = (const float*)d_in[2];
  const float* bn_w     = (const float*)d_in[3];
  const float* bn_b     = (const float*)d_in[4];
  const float* W        = (const float*)d_in[5];
  const float* att_src  = (const float*)d_in[6];
  const float* att_dst  = (const float*)d_in[7];
  const float* att_edge = (const float*)d_in[8];
  const float* W_edge   = (const float*)d_in[9];
  const float* bias     = (const float*)d_in[10];
  const float* w1 = (const float*)d_in[11];
  const float* b1 = (const float*)d_in[12];
  const float* w2 = (const float*)d_in[13];
  const float* b2 = (const float*)d_in[14];
  const float* w3 = (const float*)d_in[15];
  const float* b3 = (const float*)d_in[16];

  float* ws      = (float*)d_ws;
  float* xp      = ws + WS_XP;
  float* a_s     = ws + WS_AS;
  float* a_d     = ws + WS_AD;
  float* m       = ws + WS_M;
  float* den     = ws + WS_DEN;
  float* out_acc = ws + WS_OUT;
  float* ps      = ws + WS_PS;
  float* pe      = ws + WS_PE;
  float* scal    = ws + WS_SCAL;
  float* btp     = ws + WS_BTP;
  float* btm     = ws + WS_BTM;
  float* btb     = ws + WS_BTB;

  float* emb = (float*)d_out;                  // [N,10]
  float* y   = (float*)d_out + NN * F_OUTN;    // [N,10]

  const int* si = ei;
  const int* di = ei + NE;

  const int g_init  = (NN * F_OUTN + 255) / 256;
  const int g_edge  = (NE + NN + 255) / 256;
  const int g_tiles = ((NN / 16) + 7) / 8;     // 8 waves (16-node tiles) per block

  k_init<<<g_init, 256, 0, stream>>>(out_acc, m, den);
  k_bn_partial<<<NB_STAT, 256, 0, stream>>>(h, ps);
  k_ew_partial<<<NB_EW, 256, 0, stream>>>(ew, pe);
  k_finalize<<<1, 128, 0, stream>>>(ps, pe, bn_w, bn_b, W, att_src, att_dst,
                                    W_edge, att_edge, w1, b1, w2, b2, w3, b3,
                                    scal, btp, btm, btb);
  k_project<<<g_tiles, 256, 0, stream>>>(h, scal, btp, xp, a_s, a_d);
  k_edge_max<<<g_edge, 256, 0, stream>>>(si, di, ew, a_s, a_d, scal, m);
  k_edge_expsum<<<g_edge, 256, 0, stream>>>(si, di, ew, a_s, a_d, scal, m, den);
  k_edge_scatter<<<g_edge, 256, 0, stream>>>(si, di, ew, a_s, a_d, scal, m, den,
                                             xp, out_acc);
  k_embed<<<g_init, 256, 0, stream>>>(out_acc, bias, emb);
  k_mlp<<<g_tiles, 256, 0, stream>>>(out_acc, bias, btm, btb, y);
}